// GlobalmonopolyMoE_68539088110329
// MI455X (gfx1250) — compile-verified
//
#include <hip/hip_runtime.h>
#include <hip/hip_bf16.h>

// Problem constants (from the reference)
#define JN   22
#define EN   8
#define KNB  4
#define DTN  9
#define DN   64
#define BN   128
#define H1N  256
#define H2N  128
#define DIN  2304   // DT*K*D
#define DOUT 576    // DT*D

typedef float v2f __attribute__((ext_vector_type(2)));
typedef float v8f __attribute__((ext_vector_type(8)));

// ---------------------------------------------------------------------------
// 1) Gather: inp[b][j][(t*K+k)*64+d] = x[b][t][neighbors[j][k]][d]
// ---------------------------------------------------------------------------
__global__ void gather_inp_kernel(const float* __restrict__ x,
                                  const int* __restrict__ nbr,
                                  float* __restrict__ inp, int total) {
    int i = blockIdx.x * blockDim.x + threadIdx.x;
    if (i >= total) return;
    int d  = i & 63;
    int k  = (i >> 6) & 3;
    int t  = (i / 256) % DTN;
    int j  = (i / DIN) % JN;
    int b  = i / (DIN * JN);
    int nj = nbr[j * KNB + k];
    inp[i] = x[((b * DTN + t) * JN + nj) * DN + d];
}

// ---------------------------------------------------------------------------
// 2) Generic batched fp32 WMMA GEMM:  C = act(A @ W + bias)
//    One wave per block computes a 32x64 tile of C (2 M-tiles x 4 N-tiles,
//    8 v8f accumulators) with V_WMMA_F32_16X16X4_F32, K-loop step 4,
//    software-pipelined: next K-step operands are loaded before the current
//    K-step's 8 wmmas issue, so waits overlap with XDL execution.
//    Batch (gridDim.y) enumerates (j,e); j = batch>>3, e = batch&7.
// ---------------------------------------------------------------------------
__global__ void __launch_bounds__(32)
wmma_gemm_f32(const float* __restrict__ A, const float* __restrict__ W,
              const float* __restrict__ bias, float* __restrict__ C,
              int M, int N, int Kd,
              long a_row, long a_joff, long a_eoff, long c_row, int do_relu) {
    const int lane = threadIdx.x;          // 0..31, all lanes active (EXEC all-1s)
    const int half = lane >> 4;            // 0: K pair {0,1}, 1: K pair {2,3}
    const int l15  = lane & 15;

    const int tilesM = M >> 5;             // 32-row tiles
    const int tm = blockIdx.x % tilesM;
    const int tn = blockIdx.x / tilesM;
    const int m0 = tm << 5;
    const int n0 = tn << 6;                // 64-wide N slab
    const int batch = blockIdx.y;

    const long aoff = (long)(batch >> 3) * a_joff + (long)(batch & 7) * a_eoff;
    const float* Ap0 = A + aoff + (long)(m0 + l15) * a_row + 2 * half;
    const float* Ap1 = Ap0 + (long)16 * a_row;
    const float* Wp  = W + (long)batch * Kd * N + (long)(2 * half) * N + n0 + l15;

    v8f acc[2][4];
    #pragma unroll
    for (int mt = 0; mt < 2; ++mt)
        #pragma unroll
        for (int t = 0; t < 4; ++t) acc[mt][t] = {};

    v2f a_cur[2], b_cur[4], a_nxt[2], b_nxt[4];

    auto loadk = [&](int kb, v2f (&a)[2], v2f (&b)[4]) {
        // A 16x4 tiles: lanes 0-15 hold (M=l15, K=kb+0/1), lanes 16-31 K=kb+2/3
        const float2 av0 = *reinterpret_cast<const float2*>(Ap0 + kb);
        const float2 av1 = *reinterpret_cast<const float2*>(Ap1 + kb);
        a[0][0] = av0.x; a[0][1] = av0.y;
        a[1][0] = av1.x; a[1][1] = av1.y;
        // B 4x16 tiles: lanes 0-15 rows K=kb+0/1, lanes 16-31 rows K=kb+2/3
        const float* wrow = Wp + (long)kb * N;
        #pragma unroll
        for (int t = 0; t < 4; ++t) {
            b[t][0] = wrow[t * 16];
            b[t][1] = wrow[(long)N + t * 16];
        }
    };

    loadk(0, a_cur, b_cur);
    int kb = 0;
    for (; kb + 4 < Kd; kb += 4) {
        loadk(kb + 4, a_nxt, b_nxt);   // prefetch next K-step
        #pragma unroll
        for (int mt = 0; mt < 2; ++mt)
            #pragma unroll
            for (int t = 0; t < 4; ++t)
                acc[mt][t] = __builtin_amdgcn_wmma_f32_16x16x4_f32(
                    false, a_cur[mt], false, b_cur[t], (short)0,
                    acc[mt][t], false, false);
        #pragma unroll
        for (int mt = 0; mt < 2; ++mt) a_cur[mt] = a_nxt[mt];
        #pragma unroll
        for (int t = 0; t < 4; ++t) b_cur[t] = b_nxt[t];
    }
    // drain last K-step
    #pragma unroll
    for (int mt = 0; mt < 2; ++mt)
        #pragma unroll
        for (int t = 0; t < 4; ++t)
            acc[mt][t] = __builtin_amdgcn_wmma_f32_16x16x4_f32(
                false, a_cur[mt], false, b_cur[t], (short)0,
                acc[mt][t], false, false);

    // Epilogue: C/D layout = 8 VGPRs; VGPR r -> (M = r + 8*half, N = l15)
    const float* bp = bias + (long)batch * N + n0 + l15;
    float* Cp = C + (long)batch * N + n0 + l15;
    #pragma unroll
    for (int t = 0; t < 4; ++t) {
        const float bv = bp[t * 16];
        #pragma unroll
        for (int mt = 0; mt < 2; ++mt) {
            #pragma unroll
            for (int r = 0; r < 8; ++r) {
                float v = acc[mt][t][r] + bv;
                if (do_relu) v = fmaxf(v, 0.0f);
                Cp[(long)(m0 + mt * 16 + r + 8 * half) * c_row + t * 16] = v;
            }
        }
    }
}

// ---------------------------------------------------------------------------
// 3) Expert selection + output gather. One block per (b,j); 8 waves, wave e
//    computes sum((xhat[b,j,e,:]-target)^2); argmin (first-min) picks expert.
// ---------------------------------------------------------------------------
__global__ void __launch_bounds__(256)
select_kernel(const float* __restrict__ xh, const float* __restrict__ mu,
              const float* __restrict__ lv, const float* __restrict__ x,
              float* __restrict__ out_mu, float* __restrict__ out_lv,
              float* __restrict__ out_xh, int* __restrict__ out_idx) {
    const int b = blockIdx.x / JN;
    const int j = blockIdx.x % JN;
    const int tid = threadIdx.x;
    const int e = tid >> 5;
    const int lane = tid & 31;

    __shared__ float serr[EN];
    __shared__ int sbest;

    const long xbase = ((long)(b * JN + j) * EN + e) * DOUT;
    float s = 0.0f;
    for (int i = lane; i < DOUT; i += 32) {
        const int t = i >> 6, d = i & 63;
        const float tgt = x[((b * DTN + t) * JN + j) * DN + d];
        const float diff = xh[xbase + i] - tgt;
        s += diff * diff;
    }
    #pragma unroll
    for (int off = 16; off > 0; off >>= 1) s += __shfl_xor(s, off, 32);
    if (lane == 0) serr[e] = s;
    __syncthreads();

    if (tid == 0) {
        int best = 0; float bv = serr[0];
        #pragma unroll
        for (int k = 1; k < EN; ++k)
            if (serr[k] < bv) { bv = serr[k]; best = k; }
        sbest = best;
        out_idx[b * JN + j] = best;
    }
    __syncthreads();
    const int best = sbest;

    const long sel = ((long)(b * JN + j) * EN + best);
    for (int i = tid; i < H2N; i += 256) {
        out_mu[(long)(b * JN + j) * H2N + i] = mu[sel * H2N + i];
        out_lv[(long)(b * JN + j) * H2N + i] = lv[sel * H2N + i];
    }
    for (int i = tid; i < DOUT; i += 256) {
        const int t = i >> 6, d = i & 63;
        out_xh[((long)(b * DTN + t) * JN + j) * DN + d] = xh[sel * DOUT + i];
    }
}

// ---------------------------------------------------------------------------
extern "C" void kernel_launch(void* const* d_in, const int* in_sizes, int n_in,
                              void* d_out, int out_size, void* d_ws, size_t ws_size,
                              hipStream_t stream) {
    const float* x    = (const float*)d_in[0];
    const int*   nbr  = (const int*)  d_in[1];
    const float* W_e1 = (const float*)d_in[2];
    const float* b_e1 = (const float*)d_in[3];
    const float* W_e2 = (const float*)d_in[4];
    const float* b_e2 = (const float*)d_in[5];
    const float* W_mu = (const float*)d_in[6];
    const float* b_mu = (const float*)d_in[7];
    const float* W_lv = (const float*)d_in[8];
    const float* b_lv = (const float*)d_in[9];
    const float* W_d1 = (const float*)d_in[10];
    const float* b_d1 = (const float*)d_in[11];
    const float* W_d2 = (const float*)d_in[12];
    const float* b_d2 = (const float*)d_in[13];

    // Workspace layout (floats)
    float* ws  = (float*)d_ws;
    float* inp = ws;                                  // B*J*DIN
    float* h1  = inp + (size_t)BN * JN * DIN;         // B*J*E*H1
    float* h2  = h1  + (size_t)BN * JN * EN * H1N;    // B*J*E*H2
    float* mu  = h2  + (size_t)BN * JN * EN * H2N;
    float* lv  = mu  + (size_t)BN * JN * EN * H2N;
    float* g   = lv  + (size_t)BN * JN * EN * H2N;    // B*J*E*H1
    float* xh  = g   + (size_t)BN * JN * EN * H1N;    // B*J*E*DOUT

    // Output layout (concatenated flat, reference return order)
    float* out    = (float*)d_out;
    float* out_mu = out;                              // B*J*H2
    float* out_lv = out + (size_t)BN * JN * H2N;
    float* out_xh = out_lv + (size_t)BN * JN * H2N;   // B*DT*J*D
    int*   out_id = (int*)(out_xh + (size_t)BN * DTN * JN * DN);

    // 1) gather neighbor features
    const int tot = BN * JN * DIN;
    gather_inp_kernel<<<(tot + 255) / 256, 256, 0, stream>>>(x, nbr, inp, tot);

    const int NB = JN * EN;  // 176 (j,e) batches
    // 2) h1 = relu(inp @ W_e1 + b_e1)   M=128,N=256,K=2304
    wmma_gemm_f32<<<dim3((BN / 32) * (H1N / 64), NB), 32, 0, stream>>>(
        inp, W_e1, b_e1, h1, BN, H1N, DIN,
        (long)JN * DIN, (long)DIN, 0L, (long)JN * EN * H1N, 1);
    // 3) h2 = relu(h1 @ W_e2 + b_e2)    M=128,N=128,K=256
    wmma_gemm_f32<<<dim3((BN / 32) * (H2N / 64), NB), 32, 0, stream>>>(
        h1, W_e2, b_e2, h2, BN, H2N, H1N,
        (long)JN * EN * H1N, (long)EN * H1N, (long)H1N, (long)JN * EN * H2N, 1);
    // 4) mu = h2 @ W_mu + b_mu          M=128,N=128,K=128
    wmma_gemm_f32<<<dim3((BN / 32) * (H2N / 64), NB), 32, 0, stream>>>(
        h2, W_mu, b_mu, mu, BN, H2N, H2N,
        (long)JN * EN * H2N, (long)EN * H2N, (long)H2N, (long)JN * EN * H2N, 0);
    // 5) lv = h2 @ W_lv + b_lv
    wmma_gemm_f32<<<dim3((BN / 32) * (H2N / 64), NB), 32, 0, stream>>>(
        h2, W_lv, b_lv, lv, BN, H2N, H2N,
        (long)JN * EN * H2N, (long)EN * H2N, (long)H2N, (long)JN * EN * H2N, 0);
    // 6) g = relu(mu @ W_d1 + b_d1)     M=128,N=256,K=128
    wmma_gemm_f32<<<dim3((BN / 32) * (H1N / 64), NB), 32, 0, stream>>>(
        mu, W_d1, b_d1, g, BN, H1N, H2N,
        (long)JN * EN * H2N, (long)EN * H2N, (long)H2N, (long)JN * EN * H1N, 1);
    // 7) xh = g @ W_d2 + b_d2           M=128,N=576,K=256
    wmma_gemm_f32<<<dim3((BN / 32) * (DOUT / 64), NB), 32, 0, stream>>>(
        g, W_d2, b_d2, xh, BN, DOUT, H1N,
        (long)JN * EN * H1N, (long)EN * H1N, (long)H1N, (long)JN * EN * DOUT, 0);

    // 8) expert selection + output gather
    select_kernel<<<BN * JN, 256, 0, stream>>>(xh, mu, lv, x,
                                               out_mu, out_lv, out_xh, out_id);
}